// RecurrentLIFLayer_51599737094641
// MI455X (gfx1250) — compile-verified
//
#include <hip/hip_runtime.h>
#include <cstdint>
#include <cstddef>

// Problem constants (from reference setup_inputs):
//   batch=256, S=256 input events, N=128 neurons, 256 synapses, M=256 max spikes
#define BATCH 256
#define S_IN  256
#define NNEUR 128
#define NSYN  256
#define WPB   2              // waves (batch elements) per block -> each wave ~owns a SIMD32
#define TPB   (WPB * 32)     // 64 threads

// ---------------- dynamic LDS layout (bytes) ----------------
#define L_WFF   0                          // WffT staged: 256*128 f32 = 128KB
#define L_WREC  (NSYN * NNEUR * 4)         // WrecT staged: 128*128 f32 = 64KB
#define L_KEYS  (L_WREC + NNEUR * NNEUR * 4)       // WPB*256 u64 sort keys
#define L_ST    (L_KEYS + WPB * S_IN * 8)          // staged times
#define L_SS    (L_ST + WPB * S_IN * 4)            // staged sids
#define L_TOTAL (L_SS + WPB * S_IN * 4)            // = 204800 (fits 320KB/WGP, 1 block/WGP)

// ---------------- CDNA5 TDM path (probe-verified builtins) ----------------
#if defined(__HIP_DEVICE_COMPILE__) && defined(__gfx1250__)
#if __has_builtin(__builtin_amdgcn_tensor_load_to_lds) && \
    __has_builtin(__builtin_amdgcn_s_wait_tensorcnt)
#define USE_TDM 1
#endif
#endif

#if __has_include(<hip/amd_detail/amd_gfx1250_TDM.h>)
#define TDM_6ARG 1   // amdgpu-toolchain (therock-10.0 headers): 6-arg builtin
#endif

// fast hardware reciprocal (v_rcp_f32) with host-pass fallback
#if defined(__HIP_DEVICE_COMPILE__) && __has_builtin(__builtin_amdgcn_rcpf)
#define FRCP(x) __builtin_amdgcn_rcpf(x)
#else
#define FRCP(x) (1.0f / (x))
#endif

#ifdef USE_TDM
typedef unsigned int u32x4 __attribute__((ext_vector_type(4)));
typedef int          i32x4 __attribute__((ext_vector_type(4)));
typedef int          i32x8 __attribute__((ext_vector_type(8)));

__device__ __forceinline__ int rfl_i(int x) { return __builtin_amdgcn_readfirstlane(x); }

// 1-D contiguous global->LDS DMA of nelem 4-byte elements via the Tensor Data
// Mover.  D# per CDNA5 ISA:
//   group0: count=1, lds_addr, global_addr[56:0], type=2
//   group1: data_size=4B, tensor_dim0=nelem, tensor_dim1=1, tile_dim0=nelem,
//           tile_dim1/2 unused, tensor_dim0_stride=nelem
__device__ __forceinline__ void tdm_load_1d(unsigned lds_off, const void* gsrc,
                                            unsigned nelem) {
  unsigned long long ga = (unsigned long long)(uintptr_t)gsrc;
  u32x4 g0;
  g0[0] = (unsigned)rfl_i(1);                                   // count=1
  g0[1] = (unsigned)rfl_i((int)lds_off);                        // lds_addr
  g0[2] = (unsigned)rfl_i((int)(unsigned)ga);                   // gaddr[31:0]
  g0[3] = (unsigned)rfl_i((int)((((unsigned)(ga >> 32)) & 0x01FFFFFFu)
                                | 0x80000000u));                // gaddr[56:32] | type=2
  i32x8 g1;
  g1[0] = rfl_i((int)(2u << 16));                               // data_size=4B
  g1[1] = rfl_i((int)((nelem & 0xFFFFu) << 16));                // tensor_dim0 lo16
  g1[2] = rfl_i((int)(((nelem >> 16) & 0xFFFFu) | (1u << 16))); // dim0 hi16 | tensor_dim1=1
  g1[3] = rfl_i((int)((nelem & 0xFFFFu) << 16));                // tile_dim0
  g1[4] = 0;                                                    // tile_dim1/2 unused
  g1[5] = rfl_i((int)nelem);                                    // tensor_dim0_stride lo32
  g1[6] = 0;
  g1[7] = 0;
  i32x4 gz = (i32x4)0;
#ifdef TDM_6ARG
  i32x8 gz8 = (i32x8)0;
  __builtin_amdgcn_tensor_load_to_lds(g0, g1, gz, gz, gz8, 0);
#else
  __builtin_amdgcn_tensor_load_to_lds(g0, g1, gz, gz, 0);
#endif
}
#endif // USE_TDM

// ---------------- kernel 1: transpose weights into workspace ----------------
// WffT[s*128+n] = Wff[n*256+s]   (32768 floats)
// WrecT[j*128+n] = Wrec[n*128+j] (16384 floats)
__global__ void lif_transpose_kernel(const float* __restrict__ wff,
                                     const float* __restrict__ wrec,
                                     float* __restrict__ wffT,
                                     float* __restrict__ wrecT) {
  int i = blockIdx.x * blockDim.x + threadIdx.x;
  if (i < NSYN * NNEUR) {
    int s = i >> 7, n = i & (NNEUR - 1);
    wffT[i] = wff[n * NSYN + s];
  } else {
    int i2 = i - NSYN * NNEUR;
    if (i2 < NNEUR * NNEUR) {
      int j = i2 >> 7, n = i2 & (NNEUR - 1);
      wrecT[i2] = wrec[n * NNEUR + j];
    }
  }
}

// ---------------- kernel 2: initialize outputs (inf / -1) ----------------
__global__ void lif_init_out_kernel(float* __restrict__ out, int half, int total) {
  int i = blockIdx.x * blockDim.x + threadIdx.x;
  if (i >= total) return;
  if (i < half) out[i] = __builtin_inff();
  else ((int*)out)[i] = -1;
}

// ---------------- kernel 3: event-driven LIF, one wave per batch ----------------
__global__ void __launch_bounds__(TPB)
lif_event_kernel(const float* __restrict__ in_t, const int* __restrict__ in_s,
                 const float* __restrict__ wffT, const float* __restrict__ wrecT,
                 float* __restrict__ out_t, int* __restrict__ out_id, int M) {
  extern __shared__ char smem[];
  float* sWff = (float*)(smem + L_WFF);
  float* sWrec = (float*)(smem + L_WREC);
  unsigned long long* sKeys = (unsigned long long*)(smem + L_KEYS);
  float* sT = (float*)(smem + L_ST);
  int* sS = (int*)(smem + L_SS);

  const int lane = threadIdx.x & 31;
  const int w = threadIdx.x >> 5;
  const int b = blockIdx.x * WPB + w;

#ifdef USE_TDM
  // split the two weight-matrix DMAs across the block's two waves (parallel TDM)
  if (w == 0) tdm_load_1d((unsigned)(uintptr_t)(void*)sWff, wffT, (unsigned)(NSYN * NNEUR));
  if (w == 1) tdm_load_1d((unsigned)(uintptr_t)(void*)sWrec, wrecT, (unsigned)(NNEUR * NNEUR));
  // each wave DMAs its own batch's event arrays
  tdm_load_1d((unsigned)(uintptr_t)(void*)(sT + w * S_IN), in_t + b * S_IN, S_IN);
  tdm_load_1d((unsigned)(uintptr_t)(void*)(sS + w * S_IN), in_s + b * S_IN, S_IN);
  __builtin_amdgcn_s_wait_tensorcnt(0);
#else
  for (int i = threadIdx.x; i < (NSYN * NNEUR) / 4; i += TPB)
    ((float4*)sWff)[i] = ((const float4*)wffT)[i];
  for (int i = threadIdx.x; i < (NNEUR * NNEUR) / 4; i += TPB)
    ((float4*)sWrec)[i] = ((const float4*)wrecT)[i];
  for (int i = lane; i < S_IN; i += 32) {
    sT[w * S_IN + i] = in_t[b * S_IN + i];
    sS[w * S_IN + i] = in_s[b * S_IN + i];
  }
#endif
  __syncthreads();

  // pack (time,sid) into u64 sort keys (times are non-negative -> bit order == value order)
  unsigned long long* mk = sKeys + w * S_IN;
  for (int m = 0; m < S_IN / 32; ++m) {
    int i = lane + 32 * m;
    unsigned tb = __float_as_uint(sT[w * S_IN + i]);
    unsigned sv = (unsigned)sS[w * S_IN + i];
    mk[i] = ((unsigned long long)tb << 32) | sv;
  }
  __syncthreads();

  // per-wave bitonic sort (ascending); all waves share the same static barrier count
  for (int k = 2; k <= S_IN; k <<= 1) {
    for (int j = k >> 1; j > 0; j >>= 1) {
      for (int m = 0; m < S_IN / 32; ++m) {
        int i = lane + 32 * m;
        int l = i ^ j;
        if (l > i) {
          unsigned long long a = mk[i], c = mk[l];
          bool up = ((i & k) == 0);
          if ((a > c) == up) { mk[i] = c; mk[l] = a; }
        }
      }
      __syncthreads();
    }
  }

  // ---------------- sequential event loop ----------------
  const float INF = __builtin_inff();
  float A[4] = {0.f, 0.f, 0.f, 0.f};
  float Bv[4] = {0.f, 0.f, 0.f, 0.f};
  int ptr = 0, cnt = 0;
  float t_cur = 0.0f;
  const int base_n = lane * 4;  // lane owns neurons 4*lane .. 4*lane+3

  for (int it = 0; it < S_IN + M; ++it) {
    float t_in;
    int sid;
    if (ptr < S_IN) {
      unsigned long long kk = mk[ptr];
      t_in = __uint_as_float((unsigned)(kk >> 32));
      sid = (int)(unsigned)(kk & 0xFFFFFFFFull);
    } else {
      t_in = INF;
      sid = 0;
    }
    // prefetch feed-forward row early (ds_load_b128); only used on consume
    const float4 win = *(const float4*)(sWff + sid * NNEUR + base_n);

    // next threshold crossing: A z - B z^2 = 1, z = exp(-t/2).
    // t = -2 log z is monotone-decreasing in z, so argmin(t) == argmax(valid z).
    float zc = __expf(-0.5f * t_cur) * (1.0f - 1e-6f);
    float zb = -1.0f;
    int nb = 0;
#pragma unroll
    for (int q = 0; q < 4; ++q) {
      float a = A[q], bb = Bv[q];
      float disc = a * a - 4.0f * bb;                 // THETA = 1
      bool okd = (disc > 0.0f) && (bb > 1e-12f);
      float sq = sqrtf(okd ? disc : 1.0f);
      float inv = 0.5f * FRCP(okd ? bb : 1.0f);       // v_rcp_f32 on device
      float z1 = (a - sq) * inv;
      float z2 = (a + sq) * inv;
      float zn = -1.0f;
      if (okd && z1 > 1e-12f && z1 < zc && z1 > zn) zn = z1;
      if (okd && z2 > 1e-12f && z2 < zc && z2 > zn) zn = z2;
      if (zn > zb) { zb = zn; nb = base_n + q; }      // strict > keeps lowest index on ties
    }
    // wave32 argmax(z): positive-float bits order as signed ints, sentinel -1.0f
    // is a negative int.  5 int-max shuffles + ballot; lowest set lane == lowest
    // neuron index (lane L owns neurons 4L..4L+3) -> matches argmin tie-break.
    int zbits = __float_as_int(zb);
    int vmax = zbits;
    for (int off = 16; off > 0; off >>= 1) {
      int ov = __shfl_xor(vmax, off, 32);
      vmax = (ov > vmax) ? ov : vmax;
    }
    unsigned long long wmask = __ballot(zbits == vmax);
    int srcl = __ffsll(wmask) - 1;
    int j = __shfl(nb, srcl, 32);
    float zmax = __int_as_float(vmax);
    float t_spk = (zmax > 0.0f) ? (-2.0f * __logf(zmax)) : INF;  // single v_log_f32 per step

    bool emit = (t_spk < t_in) && (t_spk < 20.0f) && (cnt < M);
    bool consume = (!emit) && (t_in < INF);
    if (!(emit || consume)) break;  // state frozen forever after -> safe early exit

    float t_ev = emit ? t_spk : t_in;
    float em = __expf(0.5f * t_ev);  // exp(t/TAU_M), v_exp_f32
    float es = __expf(t_ev);         // exp(t/TAU_S), v_exp_f32

    if (consume) {
      A[0] += win.x * em; Bv[0] += win.x * es;
      A[1] += win.y * em; Bv[1] += win.y * es;
      A[2] += win.z * em; Bv[2] += win.z * es;
      A[3] += win.w * em; Bv[3] += win.w * es;
      ++ptr;
    } else {
      const float4 wr = *(const float4*)(sWrec + j * NNEUR + base_n);
      float r0 = wr.x - ((base_n + 0) == j ? 1.0f : 0.0f);  // reset-by-subtraction on A only
      float r1 = wr.y - ((base_n + 1) == j ? 1.0f : 0.0f);
      float r2 = wr.z - ((base_n + 2) == j ? 1.0f : 0.0f);
      float r3 = wr.w - ((base_n + 3) == j ? 1.0f : 0.0f);
      A[0] += r0 * em; Bv[0] += wr.x * es;
      A[1] += r1 * em; Bv[1] += wr.y * es;
      A[2] += r2 * em; Bv[2] += wr.z * es;
      A[3] += r3 * em; Bv[3] += wr.w * es;
      if (lane == 0) {
        out_t[b * M + cnt] = t_spk;
        out_id[b * M + cnt] = j;
      }
      ++cnt;
    }
    t_cur = t_ev;
  }
}

// ---------------- host launcher ----------------
extern "C" void kernel_launch(void* const* d_in, const int* in_sizes, int n_in,
                              void* d_out, int out_size, void* d_ws, size_t ws_size,
                              hipStream_t stream) {
  (void)in_sizes; (void)n_in; (void)ws_size;
  const float* times = (const float*)d_in[0];
  const int* sids = (const int*)d_in[1];
  const float* wff = (const float*)d_in[2];
  const float* wrec = (const float*)d_in[3];

  const int M = out_size / (2 * BATCH);  // 256 for this harness
  float* wffT = (float*)d_ws;                       // 32768 floats
  float* wrecT = wffT + NSYN * NNEUR;               // 16384 floats
  float* out_t = (float*)d_out;
  int* out_id = (int*)d_out + BATCH * M;

  // allow 200KB dynamic LDS (ISA: up to 320KB per workgroup on CDNA5)
  (void)hipFuncSetAttribute((const void*)lif_event_kernel,
                            hipFuncAttributeMaxDynamicSharedMemorySize, L_TOTAL);

  int ntr = NSYN * NNEUR + NNEUR * NNEUR;
  lif_transpose_kernel<<<(ntr + 255) / 256, 256, 0, stream>>>(wff, wrec, wffT, wrecT);
  lif_init_out_kernel<<<(out_size + 255) / 256, 256, 0, stream>>>((float*)d_out,
                                                                  BATCH * M, out_size);
  lif_event_kernel<<<BATCH / WPB, TPB, L_TOTAL, stream>>>(times, sids, wffT, wrecT,
                                                          out_t, out_id, M);
}